// GraphTransformer_75462575391232
// MI455X (gfx1250) — compile-verified
//
#include <hip/hip_runtime.h>
#include <hip/hip_bf16.h>

typedef __attribute__((ext_vector_type(2))) float v2f;
typedef __attribute__((ext_vector_type(8))) float v8f;

// ---- monotonic float<->uint order encoding (for atomicMax on floats) ----
__device__ __forceinline__ unsigned enc_ord(float f) {
    unsigned b = __float_as_uint(f);
    return (b & 0x80000000u) ? ~b : (b | 0x80000000u);
}
// decode; returns raw bits (caller checks exponent for inf/nan)
__device__ __forceinline__ unsigned dec_ord_bits(unsigned u) {
    return (u & 0x80000000u) ? (u ^ 0x80000000u) : ~u;
}

// ---------------------------------------------------------------
// Kernel 0: zero the atomic scratch (amax, denom, acc) each call
// ---------------------------------------------------------------
__global__ void zero_ws_kernel(unsigned* __restrict__ p, size_t count) {
    size_t i = (size_t)blockIdx.x * blockDim.x + threadIdx.x;
    size_t stride = (size_t)gridDim.x * blockDim.x;
    for (; i < count; i += stride) p[i] = 0u;
}

// ---------------------------------------------------------------
// Kernel 1: fused q,k,v,skip GEMM  qkvs[N,256] = x[N,128] @ [Wq|Wk|Wv|Wsk] + bias
// 512 threads = 16 waves; wave w computes a 16x16 tile (16 nodes x 16 cols)
// using V_WMMA_F32_16X16X4_F32, K=128 -> 32 steps.
// ---------------------------------------------------------------
__global__ void __launch_bounds__(512)
node_gemm_kernel(const float* __restrict__ x,
                 const float* __restrict__ Wq, const float* __restrict__ bq,
                 const float* __restrict__ Wk, const float* __restrict__ bk,
                 const float* __restrict__ Wv, const float* __restrict__ bv,
                 const float* __restrict__ Wsk, const float* __restrict__ bsk,
                 float* __restrict__ qkvs, int N_) {
    __shared__ float sx[16 * 132];             // 16 rows x 128 cols, pad to 132 (bank spread)
    const int tid = threadIdx.x;
    const int n0 = blockIdx.x * 16;

    {   // cooperative coalesced load of the 16x128 x-tile
        int row = tid >> 5;                    // 0..15
        int col = (tid & 31) * 4;              // 0..124
        int rg = n0 + row; if (rg >= N_) rg = N_ - 1;
        float4 v = *(const float4*)(x + (size_t)rg * 128 + col);
        *(float4*)(&sx[row * 132 + col]) = v;
    }
    __syncthreads();

    const int wave = tid >> 5;                 // 0..15 = 16-col tile of the 256 outputs
    const int lane = tid & 31;
    const int grp  = lane >> 4;                // 0 or 1
    const int lm   = lane & 15;

    const float* W; const float* B;
    switch (wave >> 2) {
        case 0:  W = Wq;  B = bq;  break;
        case 1:  W = Wk;  B = bk;  break;
        case 2:  W = Wv;  B = bv;  break;
        default: W = Wsk; B = bsk; break;
    }
    const int col = (wave & 3) * 16 + lm;      // column within this 64-wide W

    v8f c = {};
    #pragma unroll
    for (int k = 0; k < 128; k += 4) {
        // A 16x4 f32: lanes 0-15 hold K={k,k+1}, lanes 16-31 hold K={k+2,k+3}
        v2f a, b;
        a[0] = sx[lm * 132 + k + 2 * grp];
        a[1] = sx[lm * 132 + k + 2 * grp + 1];
        // B 4x16 f32 (row striped across lanes, mirrored grouping)
        b[0] = W[(size_t)(k + 2 * grp) * 64 + col];
        b[1] = W[(size_t)(k + 2 * grp + 1) * 64 + col];
        c = __builtin_amdgcn_wmma_f32_16x16x4_f32(false, a, false, b,
                                                  (short)0, c, false, false);
    }

    const float bias = B[col];
    const int gcol = wave * 16 + lm;           // 0..255 column in qkvs
    #pragma unroll
    for (int r = 0; r < 8; ++r) {              // C/D: VGPR r -> rows r (grp0) / r+8 (grp1)
        int row = n0 + r + 8 * grp;
        if (row < N_) qkvs[(size_t)row * 256 + gcol] = c[r] + bias;
    }
}

// ---------------------------------------------------------------
// Kernel 2: per-edge alpha = dot(q_i, k_j + e)/sqrt(D); atomicMax amax[dst]
// e recomputed on the fly from 4KB We held in LDS (never materialized).
// ---------------------------------------------------------------
__global__ void __launch_bounds__(256)
edge_alpha_kernel(const int* __restrict__ ei, const float* __restrict__ ea,
                  const float* __restrict__ We, const float* __restrict__ qkvs,
                  float* __restrict__ alpha, unsigned* __restrict__ amax, int E_) {
    __shared__ float sWe[16 * 64];
    for (int i = threadIdx.x; i < 1024; i += blockDim.x) sWe[i] = We[i];
    __syncthreads();

    const int e = blockIdx.x * blockDim.x + threadIdx.x;
    if (e >= E_) return;
    const int s = ei[e];
    const int d = ei[E_ + e];

    const float4* qp = (const float4*)(qkvs + (size_t)d * 256);        // q rows
    const float4* kp = (const float4*)(qkvs + (size_t)s * 256 + 64);   // k rows

    float q[64];
    float ah[4] = {0.f, 0.f, 0.f, 0.f};
    #pragma unroll
    for (int j = 0; j < 16; ++j) {
        float4 qv = qp[j];
        q[4*j+0] = qv.x; q[4*j+1] = qv.y; q[4*j+2] = qv.z; q[4*j+3] = qv.w;
        float4 kv = kp[j];
        int h = j >> 2;
        ah[h] += qv.x*kv.x + qv.y*kv.y + qv.z*kv.z + qv.w*kv.w;
    }

    float av[16];
    const float4* eap = (const float4*)(ea + (size_t)e * 16);
    #pragma unroll
    for (int j = 0; j < 4; ++j) {
        float4 t = eap[j];
        av[4*j+0] = t.x; av[4*j+1] = t.y; av[4*j+2] = t.z; av[4*j+3] = t.w;
    }

    #pragma unroll
    for (int ci = 0; ci < 16; ++ci) {
        float a = av[ci];
        const float4* wr = (const float4*)(&sWe[ci * 64]);
        #pragma unroll
        for (int j = 0; j < 16; ++j) {
            float4 w = wr[j];
            int h = j >> 2;
            ah[h] += a * (q[4*j]*w.x + q[4*j+1]*w.y + q[4*j+2]*w.z + q[4*j+3]*w.w);
        }
    }

    #pragma unroll
    for (int h = 0; h < 4; ++h) {
        float al = ah[h] * 0.25f;                       // 1/sqrt(16)
        alpha[(size_t)e * 4 + h] = al;
        atomicMax(&amax[(size_t)d * 4 + h], enc_ord(al));
    }
}

// ---------------------------------------------------------------
// Kernel 3: ex = exp(alpha - amax[dst]); atomicAdd denom; atomicAdd ex*(v_j+e)
// into acc (unnormalized); stash ex into the alpha_sm output slot.
// ---------------------------------------------------------------
__global__ void __launch_bounds__(256)
edge_accum_kernel(const int* __restrict__ ei, const float* __restrict__ ea,
                  const float* __restrict__ We, const float* __restrict__ qkvs,
                  const float* __restrict__ alpha, const unsigned* __restrict__ amax,
                  float* __restrict__ denom, float* __restrict__ acc,
                  float* __restrict__ exout, int E_) {
    __shared__ float sWe[16 * 64];
    for (int i = threadIdx.x; i < 1024; i += blockDim.x) sWe[i] = We[i];
    __syncthreads();

    const int e = blockIdx.x * blockDim.x + threadIdx.x;
    if (e >= E_) return;
    const int s = ei[e];
    const int d = ei[E_ + e];

    float av[16];
    const float4* eap = (const float4*)(ea + (size_t)e * 16);
    #pragma unroll
    for (int j = 0; j < 4; ++j) {
        float4 t = eap[j];
        av[4*j+0] = t.x; av[4*j+1] = t.y; av[4*j+2] = t.z; av[4*j+3] = t.w;
    }

    float ev[64];
    #pragma unroll
    for (int j = 0; j < 64; ++j) ev[j] = 0.f;
    #pragma unroll
    for (int ci = 0; ci < 16; ++ci) {
        float a = av[ci];
        const float4* wr = (const float4*)(&sWe[ci * 64]);
        #pragma unroll
        for (int j = 0; j < 16; ++j) {
            float4 w = wr[j];
            ev[4*j+0] += a * w.x; ev[4*j+1] += a * w.y;
            ev[4*j+2] += a * w.z; ev[4*j+3] += a * w.w;
        }
    }

    float exh[4];
    #pragma unroll
    for (int h = 0; h < 4; ++h) {
        unsigned b = dec_ord_bits(amax[(size_t)d * 4 + h]);
        float m = __uint_as_float(b);
        if (((b >> 23) & 0xFFu) == 0xFFu) m = 0.f;      // non-finite (incl. untouched) -> 0
        float ex = __expf(alpha[(size_t)e * 4 + h] - m);
        exh[h] = ex;
        exout[(size_t)e * 4 + h] = ex;                  // normalized later
        atomicAdd(&denom[(size_t)d * 4 + h], ex);
    }

    const float* vp = qkvs + (size_t)s * 256 + 128;     // v rows
    #pragma unroll
    for (int j = 0; j < 64; ++j) {
        int h = j >> 4;
        atomicAdd(&acc[(size_t)d * 64 + j], exh[h] * (vp[j] + ev[j]));
    }
}

// ---------------------------------------------------------------
// Kernel 4: out = acc/(denom+1e-16) + skip
// ---------------------------------------------------------------
__global__ void node_out_kernel(const float* __restrict__ qkvs,
                                const float* __restrict__ acc,
                                const float* __restrict__ denom,
                                float* __restrict__ out, int N_) {
    int i = blockIdx.x * blockDim.x + threadIdx.x;
    if (i >= N_ * 64) return;
    int n = i >> 6, ci = i & 63, h = ci >> 4;
    float dn = denom[(size_t)n * 4 + h] + 1e-16f;
    out[i] = acc[i] / dn + qkvs[(size_t)n * 256 + 192 + ci];
}

// ---------------------------------------------------------------
// Kernel 5: alpha_sm = ex/(denom[dst]+1e-16) in place; bit-copy edge_index
// ---------------------------------------------------------------
__global__ void edge_final_kernel(const int* __restrict__ ei,
                                  const float* __restrict__ denom,
                                  float* __restrict__ asm_out,
                                  int* __restrict__ eidx_out, int E_) {
    size_t stride = (size_t)gridDim.x * blockDim.x;
    size_t g = (size_t)blockIdx.x * blockDim.x + threadIdx.x;
    for (size_t i = g; i < (size_t)E_ * 4; i += stride) {
        int e = (int)(i >> 2), h = (int)(i & 3);
        int d = ei[E_ + e];
        asm_out[i] = asm_out[i] / (denom[(size_t)d * 4 + h] + 1e-16f);
    }
    for (size_t i = g; i < (size_t)2 * E_; i += stride) eidx_out[i] = ei[i];
}

extern "C" void kernel_launch(void* const* d_in, const int* in_sizes, int n_in,
                              void* d_out, int out_size, void* d_ws, size_t ws_size,
                              hipStream_t stream) {
    const float* x   = (const float*)d_in[0];
    const int*   ei  = (const int*)d_in[1];
    const float* ea  = (const float*)d_in[2];
    const float* Wq  = (const float*)d_in[3];
    const float* bq  = (const float*)d_in[4];
    const float* Wk  = (const float*)d_in[5];
    const float* bk  = (const float*)d_in[6];
    const float* Wv  = (const float*)d_in[7];
    const float* bv  = (const float*)d_in[8];
    const float* We  = (const float*)d_in[9];
    const float* Wsk = (const float*)d_in[10];
    const float* bsk = (const float*)d_in[11];

    const int N_ = in_sizes[0] / 128;   // 50000
    const int E_ = in_sizes[1] / 2;     // 800000

    // workspace layout (floats): qkvs[N,256] | alpha[E,4] | amax[N,4](u32) | denom[N,4] | acc[N,64]
    float*    qkvs  = (float*)d_ws;
    float*    alpha = qkvs + (size_t)N_ * 256;
    unsigned* amax  = (unsigned*)(alpha + (size_t)E_ * 4);
    float*    denom = (float*)(amax + (size_t)N_ * 4);
    float*    acc   = denom + (size_t)N_ * 4;

    // output tuple: out[N,64] | edge_index[2,E] (int bits) | alpha_sm[E,4]
    float* out      = (float*)d_out;
    int*   eidx_out = (int*)(out + (size_t)N_ * 64);
    float* asm_out  = out + (size_t)N_ * 64 + (size_t)2 * E_;

    size_t zc = (size_t)N_ * 72;   // amax + denom + acc (all zero-bit init)
    zero_ws_kernel<<<(unsigned)((zc + 255) / 256), 256, 0, stream>>>(amax, zc);

    node_gemm_kernel<<<(N_ + 15) / 16, 512, 0, stream>>>(
        x, Wq, bq, Wk, bk, Wv, bv, Wsk, bsk, qkvs, N_);

    edge_alpha_kernel<<<(E_ + 255) / 256, 256, 0, stream>>>(
        ei, ea, We, qkvs, alpha, amax, E_);

    edge_accum_kernel<<<(E_ + 255) / 256, 256, 0, stream>>>(
        ei, ea, We, qkvs, alpha, amax, denom, acc, asm_out, E_);

    node_out_kernel<<<(N_ * 64 + 255) / 256, 256, 0, stream>>>(
        qkvs, acc, denom, out, N_);

    edge_final_kernel<<<2048, 256, 0, stream>>>(ei, denom, asm_out, eidx_out, E_);
}